// DeformCNN5LNOBN_67199058313468
// MI455X (gfx1250) — compile-verified
//
#include <hip/hip_runtime.h>
#include <cstdint>
#include <cstddef>

typedef __attribute__((ext_vector_type(2))) float v2f;
typedef __attribute__((ext_vector_type(4))) float v4f;
typedef __attribute__((ext_vector_type(8))) float v8f;

#define NEG_SLOPE 0.01f

static __device__ __forceinline__ constexpr int ilog2(int v) {
    int l = 0; while ((1 << l) < v) ++l; return l;
}

// ---------------------------------------------------------------------------
// Pack conv weights (M_REAL, CIN, 3, 3) into WMMA f32 16x16x4 A-fragment
// layout, M padded to MPAD (multiple of 16), K padded to KC*4:
// packed[((mb*KC + kc)*32 + lane)*2 + r] = W[m][kslot]
//   m = mb*16 + (lane & 15), kslot = kc*4 + (lane>>4)*2 + r.
// ---------------------------------------------------------------------------
template<int CIN, int MPAD, int M_REAL>
__global__ __launch_bounds__(256)
void pack_weights_kernel(const float* __restrict__ wt, float* __restrict__ packed)
{
    constexpr int K  = CIN * 9;
    constexpr int KC = (K + 3) / 4;
    constexpr int MB = MPAD / 16;
    unsigned idx = blockIdx.x * blockDim.x + threadIdx.x;
    if (idx >= (unsigned)(MB * KC * 64)) return;
    unsigned r    = idx & 1;
    unsigned lane = (idx >> 1) & 31;
    unsigned kc   = (idx >> 6) % KC;
    unsigned mb   = idx / (64u * KC);
    unsigned m    = mb * 16 + (lane & 15);
    unsigned ks   = kc * 4 + (lane >> 4) * 2 + r;
    float v = 0.0f;
    if (m < (unsigned)M_REAL && ks < (unsigned)K)
        v = wt[m * CIN * 9 + ks];              // (M,CIN,3,3) flat: m*CIN*9 + c*9 + tap
    packed[idx] = v;
}

// ---------------------------------------------------------------------------
// Fully fused deformable-conv layer:
//   1) stage plain 3x3 im2col tile (NPIX x K) into LDS
//   2) WMMA GEMM vs packed offset weights (18 ch padded to 32), bias folded
//      into the WMMA C operand, branch-free b128 stores into padded offt
//   3) bilinear resample im2col tile at the deformed positions -> LDS
//   4) WMMA GEMM vs packed deform weights, fused LeakyReLU, store NCHW
// One workgroup = NPIX output pixels; waves tile (MB x NB) 16x16 blocks.
// ---------------------------------------------------------------------------
template<int CIN, int COUT, int NPIX, int H, int W>
__global__ __launch_bounds__(32 * (COUT / 16) * (NPIX / 16))
void fused_layer_kernel(const float* __restrict__ in,
                        const float* __restrict__ packedOff,
                        const float* __restrict__ offBias,
                        const float* __restrict__ packedDef,
                        float* __restrict__ out)
{
    constexpr int K      = CIN * 9;
    constexpr int KC     = (K + 3) / 4;
    constexpr int KP     = KC * 4 + 2;        // even, non-64-multiple stride (bank-safe)
    constexpr int MB     = COUT / 16;
    constexpr int NB     = NPIX / 16;
    constexpr int NWAVES = MB * NB;
    constexpr int NT     = 32 * NWAVES;
    constexpr int MBO    = 2;                 // 18 offset channels -> 2 M-tiles (padded 32)
    constexpr int OP     = 36;                // offt row stride: 16B-aligned, bank-rotating
    constexpr int HW     = H * W;
    constexpr int LGW    = ilog2(W);
    constexpr int LGHW   = ilog2(HW);

    __shared__ __align__(16) float sm[NPIX][KP];   // im2col tile: [pixel][k]
    __shared__ __align__(16) float offt[NPIX][OP]; // per-pixel offset channels (32 padded)
    __shared__ __align__(16) float bb32[32];       // zero-padded offset-conv bias

    const int tid = threadIdx.x;
    const unsigned base = blockIdx.x * (unsigned)NPIX;
    const int wave = tid >> 5, lane = tid & 31;
    const int hi = lane >> 4;                 // 0: K0/K1 half, 1: K2/K3 half

    if (tid < 32) bb32[tid] = (tid < 18) ? offBias[tid] : 0.0f;

    // ---- zero the K -> KC*4 pad region (only when K % 4 != 0, i.e. layer 0) ----
    if constexpr ((K % 4) != 0) {
        constexpr int PADK = KC * 4 - K;
        for (int i = tid; i < NPIX * PADK; i += NT)
            sm[i / PADK][K + (i % PADK)] = 0.0f;
    }

    // ---- stage plain (zero-padded) 3x3 im2col tile ----
    for (int t = tid; t < NPIX * 9; t += NT) {
        int pix = t / 9, tap = t % 9;
        unsigned p = base + (unsigned)pix;
        unsigned b = p >> LGHW, rem = p & (HW - 1);
        int h = (int)(rem >> LGW), w = (int)(rem & (W - 1));
        int y = h + tap / 3 - 1, x = w + tap % 3 - 1;
        bool valid = (y >= 0) & (y < H) & (x >= 0) & (x < W);
        int yc = min(max(y, 0), H - 1), xc = min(max(x, 0), W - 1);
        unsigned ofs = (unsigned)(yc * W + xc);
        const float* xb = in + b * (unsigned)(CIN * HW);
#pragma unroll 4
        for (int c = 0; c < CIN; ++c) {
            float v = xb[(unsigned)c * HW + ofs];
            sm[pix][c * 9 + tap] = valid ? v : 0.0f;
        }
    }
    __syncthreads();

    // ---- offset conv as WMMA GEMM (M=18 padded to 32), bias as C operand ----
    for (int t2 = wave; t2 < MBO * NB; t2 += NWAVES) {
        const int mbo = t2 / NB, nb2 = t2 % NB;
        const int n2 = nb2 * 16 + (lane & 15);
        const int mbase = mbo * 16 + hi * 8;
        // init accumulator with bias (depends on M only)
        const v4f b0 = *reinterpret_cast<const v4f*>(&bb32[mbase]);
        const v4f b1 = *reinterpret_cast<const v4f*>(&bb32[mbase + 4]);
        v8f oac;
        oac[0] = b0[0]; oac[1] = b0[1]; oac[2] = b0[2]; oac[3] = b0[3];
        oac[4] = b1[0]; oac[5] = b1[1]; oac[6] = b1[2]; oac[7] = b1[3];
        const float* Ap = packedOff + (unsigned)mbo * KC * 64;
        const float* Bp = &sm[n2][0];
#pragma unroll 4
        for (int kc = 0; kc < KC; ++kc) {
            v2f a = *reinterpret_cast<const v2f*>(Ap + kc * 64 + lane * 2);
            v2f bf = *reinterpret_cast<const v2f*>(Bp + kc * 4 + hi * 2);
            oac = __builtin_amdgcn_wmma_f32_16x16x4_f32(
                false, a, false, bf, (short)0, oac, false, false);
        }
        // branch-free b128 stores into padded offt (rows 18..31 unused)
        v4f lo, hh;
        lo[0] = oac[0]; lo[1] = oac[1]; lo[2] = oac[2]; lo[3] = oac[3];
        hh[0] = oac[4]; hh[1] = oac[5]; hh[2] = oac[6]; hh[3] = oac[7];
        *reinterpret_cast<v4f*>(&offt[n2][mbase])     = lo;
        *reinterpret_cast<v4f*>(&offt[n2][mbase + 4]) = hh;
    }
    __syncthreads();

    // ---- bilinear im2col resampling at deformed positions (overwrite sm) ----
    for (int t = tid; t < NPIX * 9; t += NT) {
        int pix = t / 9, tap = t % 9;
        unsigned p = base + (unsigned)pix;
        unsigned b = p >> LGHW, rem = p & (HW - 1);
        int h = (int)(rem >> LGW), w = (int)(rem & (W - 1));
        float ys = (float)(h + tap / 3 - 1) + offt[pix][2 * tap];
        float xs = (float)(w + tap % 3 - 1) + offt[pix][2 * tap + 1];
        float yf = floorf(ys), xf = floorf(xs);
        float wy = ys - yf, wx = xs - xf;
        int y0 = (int)yf, x0 = (int)xf;
        int y1 = y0 + 1, x1 = x0 + 1;
        bool vy0 = (y0 >= 0) & (y0 < H), vy1 = (y1 >= 0) & (y1 < H);
        bool vx0 = (x0 >= 0) & (x0 < W), vx1 = (x1 >= 0) & (x1 < W);
        float w00 = (vy0 & vx0) ? (1.0f - wy) * (1.0f - wx) : 0.0f;
        float w01 = (vy0 & vx1) ? (1.0f - wy) * wx          : 0.0f;
        float w10 = (vy1 & vx0) ? wy * (1.0f - wx)          : 0.0f;
        float w11 = (vy1 & vx1) ? wy * wx                   : 0.0f;
        int y0c = min(max(y0, 0), H - 1), y1c = min(max(y1, 0), H - 1);
        int x0c = min(max(x0, 0), W - 1), x1c = min(max(x1, 0), W - 1);
        unsigned i00 = (unsigned)(y0c * W + x0c), i01 = (unsigned)(y0c * W + x1c);
        unsigned i10 = (unsigned)(y1c * W + x0c), i11 = (unsigned)(y1c * W + x1c);
        const float* xb = in + b * (unsigned)(CIN * HW);
#pragma unroll 4
        for (int c = 0; c < CIN; ++c) {
            const float* xc = xb + (unsigned)c * HW;
            float v = w00 * xc[i00] + w01 * xc[i01] + w10 * xc[i10] + w11 * xc[i11];
            sm[pix][c * 9 + tap] = v;
        }
    }
    __syncthreads();

    // ---- deform GEMM: each wave owns one 16(M) x 16(N) tile ----
    const int mb = wave / NB, nb = wave % NB;
    const int nl = nb * 16 + (lane & 15);

    v8f acc = {};
    const float* Ap = packedDef + (unsigned)mb * KC * 64;
    const float* Bp = &sm[nl][0];
#pragma unroll 4
    for (int kc = 0; kc < KC; ++kc) {
        v2f a = *reinterpret_cast<const v2f*>(Ap + kc * 64 + lane * 2);
        v2f bf = *reinterpret_cast<const v2f*>(Bp + kc * 4 + hi * 2);
        acc = __builtin_amdgcn_wmma_f32_16x16x4_f32(
            false, a, false, bf, (short)0, acc, false, false);
    }

    // ---- fused LeakyReLU + store (D layout: VGPR r -> M = r + hi*8) ----
    unsigned p = base + (unsigned)nl;
    unsigned b = p >> LGHW, rem = p & (HW - 1);
    unsigned h = rem >> LGW, w = rem & (W - 1);
    float* op = out + ((b * COUT + (unsigned)(mb * 16 + hi * 8)) << LGHW) + (h << LGW) + w;
#pragma unroll
    for (int r = 0; r < 8; ++r) {
        float v = acc[r];
        v = (v >= 0.0f) ? v : NEG_SLOPE * v;
        op[(unsigned)r << LGHW] = v;          // COUT is a multiple of 16: no guard
    }
}

// ---------------------------------------------------------------------------
// 2x2 max pool, stride 2, NCHW. H, W are input (pre-pool) dims.
// ---------------------------------------------------------------------------
template<int C, int H, int W>
__global__ __launch_bounds__(256)
void maxpool_kernel(const float* __restrict__ in, float* __restrict__ out, int B)
{
    constexpr int Ho = H / 2, Wo = W / 2;
    constexpr int LGWo = ilog2(Wo);
    constexpr int LGHoWo = ilog2(Ho * Wo);
    unsigned idx   = blockIdx.x * blockDim.x + threadIdx.x;
    unsigned total = (unsigned)B * C * Ho * Wo;
    if (idx >= total) return;
    unsigned wo = idx & (Wo - 1);
    unsigned ho = (idx >> LGWo) & (Ho - 1);
    unsigned bc = idx >> LGHoWo;              // b*C + c
    const float* ip = in + bc * (unsigned)(H * W) + ((2u * ho) << ilog2(W)) + 2u * wo;
    float v = fmaxf(fmaxf(ip[0], ip[1]), fmaxf(ip[W], ip[W + 1]));
    out[idx] = v;
}

// ---------------------------------------------------------------------------
// Host-side per-layer driver
// ---------------------------------------------------------------------------
static inline int cdiv(long long a, long long b) { return (int)((a + b - 1) / b); }

template<int CIN, int COUT, int NPIX, int H, int W>
static void run_layer(const float* in, const float* off_w, const float* off_b,
                      const float* def_w, float* packedOff, float* packedDef,
                      float* out, int B, hipStream_t stream)
{
    constexpr int K  = CIN * 9;
    constexpr int KC = (K + 3) / 4;
    constexpr int MB = COUT / 16;
    constexpr int NB = NPIX / 16;
    constexpr int NT = 32 * MB * NB;

    pack_weights_kernel<CIN, 32, 18>
        <<<cdiv(2 * KC * 64, 256), 256, 0, stream>>>(off_w, packedOff);
    pack_weights_kernel<CIN, COUT, COUT>
        <<<cdiv(MB * KC * 64, 256), 256, 0, stream>>>(def_w, packedDef);
    fused_layer_kernel<CIN, COUT, NPIX, H, W>
        <<<(int)(((long long)B * H * W) / NPIX), NT, 0, stream>>>(in, packedOff, off_b, packedDef, out);
}

extern "C" void kernel_launch(void* const* d_in, const int* in_sizes, int n_in,
                              void* d_out, int out_size, void* d_ws, size_t ws_size,
                              hipStream_t stream)
{
    const int B = 16;
    const float* x = (const float*)d_in[0];
    const float* off_w[5]; const float* off_b[5]; const float* def_w[5];
    for (int i = 0; i < 5; ++i) {
        off_w[i] = (const float*)d_in[1 + 3 * i];
        off_b[i] = (const float*)d_in[2 + 3 * i];
        def_w[i] = (const float*)d_in[3 + 3 * i];
    }

    // Scratch layout (all 256B aligned). No offset buffer needed.
    char* ws = (char*)d_ws;
    float* actP  = (float*)(ws);                          // max 16*16*128*1024*4 = 134,217,728
    float* actQ1 = (float*)(ws + 134217728);              // max 16*16*64*512*4  =  33,554,432
    float* actQ2 = (float*)(ws + 134217728 + 33554432);   // 16*32*32*256*4      =  16,777,216
    char*  pk    = ws + 134217728 + 33554432 + 16777216;
    float* pOff[5]; float* pDef[5];
    for (int i = 0; i < 5; ++i) {
        pOff[i] = (float*)(pk + (size_t)(2 * i)     * 262144);   // 256 KB slots
        pDef[i] = (float*)(pk + (size_t)(2 * i + 1) * 262144);
    }

    // Layer 0: (1 -> 16), 128x1024, pool -> 64x512
    run_layer<1, 16, 32, 128, 1024>(x, off_w[0], off_b[0], def_w[0], pOff[0], pDef[0], actP, B, stream);
    maxpool_kernel<16, 128, 1024><<<cdiv((long long)B * 16 * 64 * 512, 256), 256, 0, stream>>>(actP, actQ1, B);

    // Layer 1: (16 -> 32), 64x512, pool -> 32x256
    run_layer<16, 32, 32, 64, 512>(actQ1, off_w[1], off_b[1], def_w[1], pOff[1], pDef[1], actP, B, stream);
    maxpool_kernel<32, 64, 512><<<cdiv((long long)B * 32 * 32 * 256, 256), 256, 0, stream>>>(actP, actQ2, B);

    // Layer 2: (32 -> 48), 32x256, pool -> 16x128
    run_layer<32, 48, 32, 32, 256>(actQ2, off_w[2], off_b[2], def_w[2], pOff[2], pDef[2], actP, B, stream);
    maxpool_kernel<48, 32, 256><<<cdiv((long long)B * 48 * 16 * 128, 256), 256, 0, stream>>>(actP, actQ1, B);

    // Layer 3: (48 -> 64), 16x128, no pool
    run_layer<48, 64, 16, 16, 128>(actQ1, off_w[3], off_b[3], def_w[3], pOff[3], pDef[3], actP, B, stream);

    // Layer 4: (64 -> 80), 16x128, no pool -> d_out (16,80,16,128)
    run_layer<64, 80, 16, 16, 128>(actP, off_w[4], off_b[4], def_w[4], pOff[4], pDef[4], (float*)d_out, B, stream);
}